// Bisa_Model_attentin_14705968021751
// MI455X (gfx1250) — compile-verified
//
#include <hip/hip_runtime.h>
#include <hip/hip_bf16.h>
#include <math.h>

// Shapes from the reference
#define B_   16
#define S_   1024
#define H_   768
#define NH_  8
#define NHP_ 16   // heads padded to 16 WMMA columns (8..15 are zeros)
#define HD_  64
#define OUT_ 4

typedef float v2f __attribute__((ext_vector_type(2)));
typedef float v8f __attribute__((ext_vector_type(8)));

// ---------------------------------------------------------------------------
// Kernel 1a: q0[b, c] = X[b, 0, :] . Wq[:, c] + bq[c]          (16 x 768)
// grid 48, block 256 : blk -> (b = blk/3, column chunk = blk%3)
// ---------------------------------------------------------------------------
__global__ __launch_bounds__(256) void k_q0(const float* __restrict__ X,
                                            const float* __restrict__ Wq,
                                            const float* __restrict__ bq,
                                            float* __restrict__ q0) {
    const int b = blockIdx.x / 3;
    const int c = (blockIdx.x % 3) * 256 + threadIdx.x;
    const float* x = X + (size_t)b * S_ * H_;   // row 0 of batch b
    float acc = bq[c];
    for (int j = 0; j < H_; ++j)
        acc = fmaf(x[j], Wq[(size_t)j * H_ + c], acc);   // Wq coalesced over c
    q0[b * H_ + c] = acc;
}

// ---------------------------------------------------------------------------
// Kernel 1b: wpad[b, h, j] = sum_d q0[b, h*64+d] * Wk[j, h*64+d]  (h < 8)
//            wpad[b, h, j] = 0                                    (h >= 8)
// (bk contributes a constant per (b,h) logit row -> cancels in softmax)
// grid (6 j-chunks, 16 padded heads), block 256
// ---------------------------------------------------------------------------
__global__ __launch_bounds__(256) void k_wvec(const float* __restrict__ Wk,
                                              const float* __restrict__ q0,
                                              float* __restrict__ wpad) {
    const int h  = blockIdx.y;
    const int jc = blockIdx.x;
    if (h >= NH_) {                      // zero-fill padding columns once
        for (int i = threadIdx.x; i < B_ * 128; i += 256) {
            const int b = i >> 7, j = i & 127;
            wpad[((size_t)(b * NHP_ + h)) * H_ + jc * 128 + j] = 0.0f;
        }
        return;
    }
    __shared__ float q0s[B_ * HD_];       // 16 x 64
    __shared__ float wks[128 * 65];       // 128 rows x 64 (+1 pad: bank-conflict-free)
    for (int i = threadIdx.x; i < B_ * HD_; i += 256) {
        const int b = i >> 6, d = i & 63;
        q0s[i] = q0[b * H_ + h * HD_ + d];
    }
    for (int i = threadIdx.x; i < 128 * 64; i += 256) {
        const int j = i >> 6, d = i & 63;
        wks[j * 65 + d] = Wk[(size_t)(jc * 128 + j) * H_ + h * HD_ + d];
    }
    __syncthreads();
    for (int i = threadIdx.x; i < B_ * 128; i += 256) {
        const int b = i >> 7, j = i & 127;
        float acc = 0.f;
        #pragma unroll 8
        for (int d = 0; d < HD_; ++d)
            acc = fmaf(q0s[b * HD_ + d], wks[j * 65 + d], acc);
        wpad[((size_t)(b * NHP_ + h)) * H_ + jc * 128 + j] = acc;
    }
}

// ---------------------------------------------------------------------------
// Kernel 2: scores[b, h, k] = (X[b, k, :] . wpad[b, h, :]) / 8
// via V_WMMA_F32_16X16X4_F32: one wave per 16-row tile; heads in N=0..7,
// N=8..15 multiply against the zero-padded wpad columns (no predication!).
// A(16x4 f32): lanes 0-15 rows, VGPR0/1 = K offsets {0,1} (lanes<16) / {2,3}
// B(4x16 f32): lanes = N column, same K-offset split per lane half
// C(16x16):    VGPR v -> M = v (lanes 0-15) / v+8 (lanes 16-31), N = lane&15
// grid (16, 16) blocks of (32, 4): 64 row-tiles x 16 batches, 1 wave per tile
// Inner loop: 2x global_load_b64 + 1x v_wmma, no EXEC juggling.
// ---------------------------------------------------------------------------
__global__ __launch_bounds__(128) void k_scores_wmma(const float* __restrict__ X,
                                                     const float* __restrict__ wpad,
                                                     float* __restrict__ scores) {
    const int b    = blockIdx.y;
    const int tile = blockIdx.x * blockDim.y + threadIdx.y;   // 0..63
    const int lane = threadIdx.x;                             // wave32
    const int row0 = tile * 16;
    const int m    = lane & 15;                 // A row / C column index
    const int kh   = (lane >> 4) << 1;          // K sub-offset: 0 or 2
    const float* xrow = X    + ((size_t)(b * S_   + row0 + m)) * H_ + kh;
    const float* wrow = wpad + ((size_t)(b * NHP_ + m))        * H_ + kh;

    v8f c = {};
    #pragma unroll 8
    for (int k = 0; k < H_; k += 4) {
        const v2f a  = *(const v2f*)(xrow + k);   // 8B-aligned -> global_load_b64
        const v2f bb = *(const v2f*)(wrow + k);   // zero columns for m >= 8
        c = __builtin_amdgcn_wmma_f32_16x16x4_f32(
                /*neg_a=*/false, a, /*neg_b=*/false, bb,
                /*c_mod=*/(short)0, c, /*reuse_a=*/false, /*reuse_b=*/false);
    }

    if (m < NH_) {
        const int mbase = row0 + ((lane >> 4) << 3);   // +8 rows for upper lanes
        float* srow = scores + ((size_t)(b * NH_ + m)) * S_;
        #pragma unroll
        for (int v = 0; v < 8; ++v)
            srow[mbase + v] = c[v] * 0.125f;           // 1/sqrt(64)
    }
}

// ---------------------------------------------------------------------------
// Kernel 3: in-place masked softmax over k (1024) per (b,h) row
// grid 128 (= B*NH), block 256, 4 elems/thread
// ---------------------------------------------------------------------------
__global__ __launch_bounds__(256) void k_softmax(float* __restrict__ scores,
                                                 const float* __restrict__ mask) {
    const int bh = blockIdx.x;
    const int b  = bh >> 3;
    float* row = scores + (size_t)bh * S_;
    const float* mrow = mask + (size_t)b * S_;
    __shared__ float red[256];
    const int t = threadIdx.x;

    float vals[4];
    float mx = -INFINITY;
    #pragma unroll
    for (int i = 0; i < 4; ++i) {
        const int k = t + i * 256;
        float s = row[k] + (1.0f - mrow[k]) * -10000.0f;
        vals[i] = s;
        mx = fmaxf(mx, s);
    }
    red[t] = mx; __syncthreads();
    for (int off = 128; off > 0; off >>= 1) {
        if (t < off) red[t] = fmaxf(red[t], red[t + off]);
        __syncthreads();
    }
    mx = red[0]; __syncthreads();

    float sum = 0.f;
    #pragma unroll
    for (int i = 0; i < 4; ++i) { vals[i] = expf(vals[i] - mx); sum += vals[i]; }
    red[t] = sum; __syncthreads();
    for (int off = 128; off > 0; off >>= 1) {
        if (t < off) red[t] += red[t + off];
        __syncthreads();
    }
    const float inv = 1.0f / red[0];
    #pragma unroll
    for (int i = 0; i < 4; ++i) row[t + i * 256] = vals[i] * inv;
}

// ---------------------------------------------------------------------------
// Kernel 4: deterministic K-split partials of t[b,h,:] = probs[b,h,:] @ X[b]
// grid (3 j-chunks, 8 k-splits, 16 batches), block 256; coalesced X stream
// ---------------------------------------------------------------------------
__global__ __launch_bounds__(256) void k_pv_partial(const float* __restrict__ X,
                                                    const float* __restrict__ probs,
                                                    float* __restrict__ tpart) {
    const int jc = blockIdx.x, ks = blockIdx.y, b = blockIdx.z;
    const int j  = jc * 256 + threadIdx.x;
    const int k0 = ks * 128;
    __shared__ float p[NH_][128];
    for (int i = threadIdx.x; i < NH_ * 128; i += 256) {
        const int h = i >> 7, kk = i & 127;
        p[h][kk] = probs[((size_t)(b * NH_ + h)) * S_ + k0 + kk];
    }
    __syncthreads();
    float acc[NH_] = {};
    const float* xp = X + ((size_t)(b * S_ + k0)) * H_ + j;
    for (int kk = 0; kk < 128; ++kk) {
        const float x = xp[(size_t)kk * H_];
        #pragma unroll
        for (int h = 0; h < NH_; ++h) acc[h] = fmaf(p[h][kk], x, acc[h]);
    }
    #pragma unroll
    for (int h = 0; h < NH_; ++h)
        tpart[(((size_t)(b * 8 + ks)) * NH_ + h) * H_ + j] = acc[h];
}

// Kernel 4b: reduce 8 K-split partials -> t[b,h,j]  (fixed order: deterministic)
__global__ __launch_bounds__(256) void k_pv_reduce(const float* __restrict__ tpart,
                                                   float* __restrict__ t) {
    const int idx = blockIdx.x * 256 + threadIdx.x;   // < 16*8*768
    const int b = idx / (NH_ * H_);
    const int r = idx % (NH_ * H_);
    const int h = r / H_;
    const int j = r % H_;
    float s = 0.f;
    #pragma unroll
    for (int ks = 0; ks < 8; ++ks)
        s += tpart[(((size_t)(b * 8 + ks)) * NH_ + h) * H_ + j];
    t[idx] = s;
}

// ---------------------------------------------------------------------------
// Kernel 5: ctx[b, c] = t[b, c/64, :] . Wv[:, c] + bv[c]    (probs sum to 1)
// grid 48, block 256
// ---------------------------------------------------------------------------
__global__ __launch_bounds__(256) void k_ctx(const float* __restrict__ t,
                                             const float* __restrict__ Wv,
                                             const float* __restrict__ bv,
                                             float* __restrict__ ctx) {
    const int b = blockIdx.x / 3;
    const int c = (blockIdx.x % 3) * 256 + threadIdx.x;
    const int h = c >> 6;
    const float* tp = t + ((size_t)(b * NH_ + h)) * H_;
    float acc = bv[c];
    for (int j = 0; j < H_; ++j)
        acc = fmaf(tp[j], Wv[(size_t)j * H_ + c], acc);
    ctx[b * H_ + c] = acc;
}

// Kernel 6: out[b, o] = relu(ctx[b,:] . Wo[:,o] + bo[o])    (16 x 4)
__global__ __launch_bounds__(64) void k_out(const float* __restrict__ ctx,
                                            const float* __restrict__ Wo,
                                            const float* __restrict__ bo,
                                            float* __restrict__ out) {
    const int tid = threadIdx.x;           // 64 threads
    const int b = tid >> 2, o = tid & 3;
    float acc = bo[o];
    const float* cp = ctx + b * H_;
    for (int c = 0; c < H_; ++c)
        acc = fmaf(cp[c], Wo[c * OUT_ + o], acc);
    out[tid] = fmaxf(acc, 0.0f);
}

// ---------------------------------------------------------------------------
extern "C" void kernel_launch(void* const* d_in, const int* in_sizes, int n_in,
                              void* d_out, int out_size, void* d_ws, size_t ws_size,
                              hipStream_t stream) {
    const float* X    = (const float*)d_in[0];   // (16,1024,768)
    const float* mask = (const float*)d_in[1];   // (16,1024)
    const float* Wq   = (const float*)d_in[2];
    const float* bq   = (const float*)d_in[3];
    const float* Wk   = (const float*)d_in[4];
    // d_in[5] = bk: per-(b,h) constant logit shift -> cancelled by softmax
    const float* Wv   = (const float*)d_in[6];
    const float* bv   = (const float*)d_in[7];
    const float* Wo   = (const float*)d_in[8];
    const float* bo   = (const float*)d_in[9];
    float* out = (float*)d_out;

    // fp32 workspace layout (offsets in floats), total ~5 MB
    float* ws     = (float*)d_ws;
    float* q0     = ws;                                  // 16*768       = 12288
    float* wpad   = q0 + B_ * H_;                        // 16*16*768    = 196608
    float* scores = wpad + (size_t)B_ * NHP_ * H_;       // 16*8*1024    = 131072
    float* tpart  = scores + (size_t)B_ * NH_ * S_;      // 16*8*8*768   = 786432
    float* t      = tpart + (size_t)B_ * 8 * NH_ * H_;   // 16*8*768     = 98304
    float* ctx    = t + B_ * NH_ * H_;                   // 16*768       = 12288
    (void)in_sizes; (void)n_in; (void)out_size; (void)ws_size;

    k_q0<<<48, 256, 0, stream>>>(X, Wq, bq, q0);
    k_wvec<<<dim3(6, NHP_), 256, 0, stream>>>(Wk, q0, wpad);
    k_scores_wmma<<<dim3(16, 16), dim3(32, 4), 0, stream>>>(X, wpad, scores);
    k_softmax<<<B_ * NH_, 256, 0, stream>>>(scores, mask);
    k_pv_partial<<<dim3(3, 8, 16), 256, 0, stream>>>(X, scores, tpart);
    k_pv_reduce<<<(B_ * NH_ * H_) / 256, 256, 0, stream>>>(tpart, t);
    k_ctx<<<48, 256, 0, stream>>>(t, Wv, bv, ctx);
    k_out<<<1, 64, 0, stream>>>(ctx, Wo, bo, out);
}